// GNN_Agent_34522947126002
// MI455X (gfx1250) — compile-verified
//
#include <hip/hip_runtime.h>
#include <hip/hip_bf16.h>
#include <math.h>

// ---------------------------------------------------------------------------
// Problem constants (from reference)
// ---------------------------------------------------------------------------
#define N_NODES  50000
#define N_FEAT   63
#define N_EDGES  800000
#define N_LABELS 10000
#define BATCH    4
#define N_LAYERS 5
#define C        64            // N_FEAT + 1
#define ROWS     (BATCH * N_NODES)        // 200000 (divisible by 16)
#define TEN      ((size_t)ROWS * C)       // 12,800,000 floats per tensor
#define LDSP     (C + 4)                  // padded LDS row stride (bank-conflict-free)
#define EPW      8                        // edges per wave in scatter

typedef float v2f __attribute__((ext_vector_type(2)));
typedef float v8f __attribute__((ext_vector_type(8)));

// CDNA5 native fp32 WMMA: D(16x16) = A(16x4) * B(4x16) + C
__device__ __forceinline__ v8f wmma_f32(v2f a, v2f b, v8f c) {
    return __builtin_amdgcn_wmma_f32_16x16x4_f32(
        /*neg_a=*/false, a, /*neg_b=*/false, b,
        /*c_mod=*/(short)0, c, /*reuse_a=*/false, /*reuse_b=*/false);
}

__device__ __forceinline__ float sigmoidf_(float v) {
    return 1.0f / (1.0f + __expf(-v));
}

// ---------------------------------------------------------------------------
// Zero fill (float4)
// ---------------------------------------------------------------------------
__global__ void zero4_kernel(float4* __restrict__ p, size_t n4) {
    size_t i = (size_t)blockIdx.x * blockDim.x + threadIdx.x;
    if (i < n4) p[i] = make_float4(0.f, 0.f, 0.f, 0.f);
}

// ---------------------------------------------------------------------------
// Build x0 = concat(nodes, 0) broadcast over batch; also initialize x = x0.
// ---------------------------------------------------------------------------
__global__ void build_x0_kernel(const float* __restrict__ nodes,
                                float* __restrict__ x0, float* __restrict__ x) {
    size_t idx = (size_t)blockIdx.x * blockDim.x + threadIdx.x;
    if (idx >= TEN) return;
    int f = (int)(idx & (C - 1));
    size_t bn = idx >> 6;                 // C == 64
    int node = (int)(bn % N_NODES);
    float v = (f < N_FEAT) ? nodes[(size_t)node * N_FEAT + f] : 0.0f;
    x0[idx] = v;
    x[idx]  = v;
}

// Scatter cov into feature 63 of labelled nodes (x0 and x).
__global__ void cov_scatter_kernel(const float* __restrict__ cov,
                                   const int* __restrict__ c2l,
                                   float* __restrict__ x0, float* __restrict__ x) {
    int i = blockIdx.x * blockDim.x + threadIdx.x;   // BATCH * N_LABELS
    if (i >= BATCH * N_LABELS) return;
    int b = i / N_LABELS, j = i % N_LABELS;
    int node = c2l[j];
    size_t off = ((size_t)b * N_NODES + node) * C + (C - 1);
    float v = cov[(size_t)b * N_LABELS + j];
    x0[off] = v;
    x[off]  = v;
}

// ---------------------------------------------------------------------------
// m = x @ W   (W: 64x64 row-major, shared across batch)
// One wave computes a 16x64 tile via 4 fp32 WMMA accumulators.
// W staged TRANSPOSED + padded in LDS: B frag = one conflict-free ds_load_b64.
// ---------------------------------------------------------------------------
__global__ void ggc_gemm_kernel(const float* __restrict__ x,
                                const float* __restrict__ W,
                                float* __restrict__ m, int rowTiles) {
    __shared__ float lWt[C * LDSP];       // lWt[j*LDSP + k] = W[k*C + j]
    for (int i = threadIdx.x; i < C * C; i += blockDim.x) {
        int k = i >> 6, j = i & (C - 1);
        lWt[j * LDSP + k] = W[i];
    }
    __syncthreads();

    int wave = (int)((blockIdx.x * blockDim.x + threadIdx.x) >> 5);
    int lane = threadIdx.x & 31;
    if (wave >= rowTiles) return;          // uniform per wave

    int row0 = wave * 16;
    int Mi = lane & 15;
    int kh = lane >> 4;                    // 0: K=k0,k0+1  1: K=k0+2,k0+3
    const float* xrow = x + ((size_t)row0 + Mi) * C;

    v8f acc[4] = {};
#pragma unroll
    for (int k0 = 0; k0 < C; k0 += 4) {
        int kk = k0 + 2 * kh;
        v2f a = *(const v2f*)&xrow[kk];
#pragma unroll
        for (int jt = 0; jt < 4; ++jt) {
            int col = jt * 16 + Mi;
            v2f bf = *(const v2f*)&lWt[col * LDSP + kk];
            acc[jt] = wmma_f32(a, bf, acc[jt]);
        }
    }
#pragma unroll
    for (int jt = 0; jt < 4; ++jt)
#pragma unroll
        for (int v = 0; v < 8; ++v)
            m[((size_t)row0 + kh * 8 + v) * C + jt * 16 + Mi] = acc[jt][v];
}

// ---------------------------------------------------------------------------
// agg[b][dst[e]] += m[b][src[e]]  — one wave per EPW-edge strip per batch,
// float2 per lane, prefetch 2 edges ahead to cover the random-row gather.
// ---------------------------------------------------------------------------
__global__ void scatter_add_kernel(const float* __restrict__ m,
                                   const int* __restrict__ src,
                                   const int* __restrict__ dst,
                                   float* __restrict__ agg) {
    int wave = (int)((blockIdx.x * blockDim.x + threadIdx.x) >> 5);
    int lane = threadIdx.x & 31;
    int b = blockIdx.y;
    int e0 = wave * EPW;
    const float* mb = m   + (size_t)b * N_NODES * C;
    float*       ab = agg + (size_t)b * N_NODES * C;

#pragma unroll
    for (int i = 0; i < EPW; ++i) {
        int e = e0 + i;                    // uniform per wave
        if (e >= N_EDGES) break;
        int ep = e + 2;
        if (ep < N_EDGES)                  // global_prefetch_b8, per-lane slice
            __builtin_prefetch((const char*)(mb + (size_t)src[ep] * C) + lane * 8, 0, 0);
        const float2* ms = (const float2*)(mb + (size_t)src[e] * C);
        float2 v = ms[lane];
        float* a = (float*)((float2*)(ab + (size_t)dst[e] * C) + lane);
        atomicAdd(a,     v.x);
        atomicAdd(a + 1, v.y);
    }
}

// ---------------------------------------------------------------------------
// Fused GRU cell: x_new = GRU(agg, h).  One wave per 16-node x 16-col tile.
// Six fp32 WMMA accumulators: (gx, gh) for each of the r, z, n gate blocks.
// w_ih / w_hh staged in dynamic LDS with padded rows (conflict-free b64).
// ---------------------------------------------------------------------------
__global__ void gru_kernel(const float* __restrict__ agg,
                           const float* __restrict__ h,
                           const float* __restrict__ w_ih,  // (3C, C)
                           const float* __restrict__ w_hh,  // (3C, C)
                           const float* __restrict__ b_ih,
                           const float* __restrict__ b_hh,
                           float* __restrict__ xout, int rowTiles) {
    extern __shared__ float lds[];
    float* lw_ih = lds;                    // 3C * LDSP
    float* lw_hh = lds + 3 * C * LDSP;
    for (int i = threadIdx.x; i < 3 * C * C; i += blockDim.x) {
        int g = i >> 6, k = i & (C - 1);
        lw_ih[g * LDSP + k] = w_ih[i];
        lw_hh[g * LDSP + k] = w_hh[i];
    }
    __syncthreads();

    int gw = (int)((blockIdx.x * blockDim.x + threadIdx.x) >> 5);
    int nWaves = rowTiles * 4;             // 4 col tiles of 16 (C = 64)
    if (gw >= nWaves) return;
    int lane = threadIdx.x & 31;
    int tile = gw >> 2;
    int j0 = (gw & 3) * 16;
    int row0 = tile * 16;
    int Mi = lane & 15;
    int kh = lane >> 4;

    const float* arow = agg + ((size_t)row0 + Mi) * C;
    const float* hrow = h   + ((size_t)row0 + Mi) * C;
    int wr = (0 * C + j0 + Mi) * LDSP;
    int wz = (1 * C + j0 + Mi) * LDSP;
    int wn = (2 * C + j0 + Mi) * LDSP;

    v8f xr = {}, hr = {}, xz = {}, hz = {}, xn = {}, hn = {};
#pragma unroll
    for (int k0 = 0; k0 < C; k0 += 4) {
        int kk = k0 + 2 * kh;
        v2f Aa = *(const v2f*)&arow[kk];
        v2f Ah = *(const v2f*)&hrow[kk];
        v2f Bir = *(const v2f*)&lw_ih[wr + kk];
        v2f Biz = *(const v2f*)&lw_ih[wz + kk];
        v2f Bin = *(const v2f*)&lw_ih[wn + kk];
        v2f Bhr = *(const v2f*)&lw_hh[wr + kk];
        v2f Bhz = *(const v2f*)&lw_hh[wz + kk];
        v2f Bhn = *(const v2f*)&lw_hh[wn + kk];
        xr = wmma_f32(Aa, Bir, xr);
        hr = wmma_f32(Ah, Bhr, hr);
        xz = wmma_f32(Aa, Biz, xz);
        hz = wmma_f32(Ah, Bhz, hz);
        xn = wmma_f32(Aa, Bin, xn);
        hn = wmma_f32(Ah, Bhn, hn);
    }

    int col = j0 + Mi;
    float bir = b_ih[col],         bhr = b_hh[col];
    float biz = b_ih[C + col],     bhz = b_hh[C + col];
    float bin = b_ih[2 * C + col], bhn = b_hh[2 * C + col];

#pragma unroll
    for (int v = 0; v < 8; ++v) {
        size_t row = (size_t)row0 + kh * 8 + v;
        float hv = h[row * C + col];
        float r = sigmoidf_(xr[v] + bir + hr[v] + bhr);
        float z = sigmoidf_(xz[v] + biz + hz[v] + bhz);
        float n = tanhf(xn[v] + bin + r * (hn[v] + bhn));
        xout[row * C + col] = (1.0f - z) * n + z * hv;
    }
}

// ---------------------------------------------------------------------------
// Attention + pool: att = sigmoid(cat@ai_w.T+ai_b) * relu(cat@aj_w.T+aj_b)
// pooled[b] += sum over nodes.  cat = [x | x0], K = 128.
// One wave per 16-node x 16-col tile; two WMMA chains; B frags from global
// (rows are k-contiguous -> coalesced global_load_b64, L2-resident).
// ---------------------------------------------------------------------------
__global__ void att_pool_kernel(const float* __restrict__ x,
                                const float* __restrict__ x0,
                                const float* __restrict__ ai_w,  // (2C,2C)
                                const float* __restrict__ ai_b,
                                const float* __restrict__ aj_w,
                                const float* __restrict__ aj_b,
                                float* __restrict__ pooled, int rowTiles) {
    int gw = (int)((blockIdx.x * blockDim.x + threadIdx.x) >> 5);
    int nWaves = rowTiles * 8;             // 8 col tiles of 16 (2C = 128)
    if (gw >= nWaves) return;
    int lane = threadIdx.x & 31;
    int tile = gw >> 3;
    int j0 = (gw & 7) * 16;
    int row0 = tile * 16;
    int Mi = lane & 15;
    int kh = lane >> 4;

    const float* xrow  = x  + ((size_t)row0 + Mi) * C;
    const float* x0row = x0 + ((size_t)row0 + Mi) * C;
    size_t wrow = ((size_t)(j0 + Mi)) * (2 * C);

    v8f aI = {}, aJ = {};
#pragma unroll
    for (int k0 = 0; k0 < 2 * C; k0 += 4) {
        int kk = k0 + 2 * kh;              // kk even; kk,kk+1 on same side of 64
        v2f A = (kk < C) ? *(const v2f*)&xrow[kk] : *(const v2f*)&x0row[kk - C];
        v2f Bi = *(const v2f*)&ai_w[wrow + kk];
        v2f Bj = *(const v2f*)&aj_w[wrow + kk];
        aI = wmma_f32(A, Bi, aI);
        aJ = wmma_f32(A, Bj, aJ);
    }

    int col = j0 + Mi;
    float bi = ai_b[col], bj = aj_b[col];
    float partial = 0.0f;
#pragma unroll
    for (int v = 0; v < 8; ++v) {
        float s  = sigmoidf_(aI[v] + bi);
        float rl = fmaxf(aJ[v] + bj, 0.0f);
        partial += s * rl;
    }
    // combine the two M-halves (lane L and L^16 hold the same column)
    partial += __shfl_xor(partial, 16, 32);
    int b = row0 / N_NODES;
    if (lane < 16) atomicAdd(&pooled[b * (2 * C) + col], partial);
}

// ---------------------------------------------------------------------------
// Head: pooled -> relu -> MLP(256) -> relu -> critic(1).  Single block.
// ---------------------------------------------------------------------------
__global__ void head_kernel(const float* __restrict__ pooled_raw,
                            const float* __restrict__ mlp_w,   // (256, 2C)
                            const float* __restrict__ mlp_b,
                            const float* __restrict__ critic_w, // (1, 256)
                            const float* __restrict__ critic_b,
                            float* __restrict__ out) {
    __shared__ float st[BATCH][256];
    int b = threadIdx.x >> 8;   // 1024 threads: 4 batches x 256 units
    int u = threadIdx.x & 255;
    float acc = mlp_b[u];
    for (int k = 0; k < 2 * C; ++k)
        acc += fmaxf(pooled_raw[b * (2 * C) + k], 0.0f) * mlp_w[(size_t)u * (2 * C) + k];
    st[b][u] = fmaxf(acc, 0.0f);
    __syncthreads();
    if (u == 0) {
        float o = critic_b[0];
        for (int k = 0; k < 256; ++k) o += st[b][k] * critic_w[k];
        out[b] = o;
    }
}

// ---------------------------------------------------------------------------
// Host side
// ---------------------------------------------------------------------------
extern "C" void kernel_launch(void* const* d_in, const int* in_sizes, int n_in,
                              void* d_out, int out_size, void* d_ws, size_t ws_size,
                              hipStream_t stream) {
    const float* cov      = (const float*)d_in[0];
    const float* nodes    = (const float*)d_in[1];
    const int*   edges    = (const int*)  d_in[2];   // (2, N_EDGES)
    const int*   c2l      = (const int*)  d_in[3];
    const float* ggc_w    = (const float*)d_in[4];   // (5, 64, 64)
    const float* gru_w_ih = (const float*)d_in[5];
    const float* gru_w_hh = (const float*)d_in[6];
    const float* gru_b_ih = (const float*)d_in[7];
    const float* gru_b_hh = (const float*)d_in[8];
    const float* ai_w     = (const float*)d_in[9];
    const float* ai_b     = (const float*)d_in[10];
    const float* aj_w     = (const float*)d_in[11];
    const float* aj_b     = (const float*)d_in[12];
    const float* mlp_w    = (const float*)d_in[13];
    const float* mlp_b    = (const float*)d_in[14];
    const float* critic_w = (const float*)d_in[15];
    const float* critic_b = (const float*)d_in[16];
    float* out = (float*)d_out;

    float* ws     = (float*)d_ws;
    float* x0     = ws;
    float* bufA   = ws + TEN;
    float* bufB   = ws + 2 * TEN;
    float* agg    = ws + 3 * TEN;
    float* pooled = ws + 4 * TEN;          // 512 floats

    const int rowTiles = ROWS / 16;        // 12500

    // init x0 and x
    {
        int blocks = (int)((TEN + 255) / 256);
        build_x0_kernel<<<blocks, 256, 0, stream>>>(nodes, x0, bufA);
        cov_scatter_kernel<<<(BATCH * N_LABELS + 255) / 256, 256, 0, stream>>>(
            cov, c2l, x0, bufA);
    }

    const int* src = edges;
    const int* dst = edges + N_EDGES;
    const size_t gruLds = (size_t)2 * 3 * C * LDSP * sizeof(float);   // ~104 KB

    float* cur = bufA;
    float* nxt = bufB;
    for (int L = 0; L < N_LAYERS; ++L) {
        // m = cur @ W[L]  -> nxt
        ggc_gemm_kernel<<<(rowTiles + 7) / 8, 256, 0, stream>>>(
            cur, ggc_w + (size_t)L * C * C, nxt, rowTiles);
        // agg = 0
        zero4_kernel<<<(int)((TEN / 4 + 255) / 256), 256, 0, stream>>>(
            (float4*)agg, TEN / 4);
        // agg[b][dst] += m[b][src]
        dim3 sgrid((N_EDGES + 8 * EPW - 1) / (8 * EPW), BATCH);  // 8 waves/block
        scatter_add_kernel<<<sgrid, 256, 0, stream>>>(nxt, src, dst, agg);
        // x_new = GRU(agg, cur) -> nxt   (m is dead now)
        int gruWaves = rowTiles * 4;
        gru_kernel<<<(gruWaves * 32 + 255) / 256, 256, gruLds, stream>>>(
            agg, cur, gru_w_ih, gru_w_hh, gru_b_ih, gru_b_hh, nxt, rowTiles);
        float* t = cur; cur = nxt; nxt = t;
    }

    // attention + pool
    zero4_kernel<<<1, 128, 0, stream>>>((float4*)pooled, (size_t)BATCH * 2 * C / 4);
    int attWaves = rowTiles * 8;
    att_pool_kernel<<<(attWaves * 32 + 255) / 256, 256, 0, stream>>>(
        cur, x0, ai_w, ai_b, aj_w, aj_b, pooled, rowTiles);

    // head
    head_kernel<<<1, 1024, 0, stream>>>(pooled, mlp_w, mlp_b, critic_w, critic_b, out);
}